// Encoder_Adversarial_GCN_21904333210049
// MI455X (gfx1250) — compile-verified
//
#include <hip/hip_runtime.h>
#include <hip/hip_bf16.h>

typedef __attribute__((ext_vector_type(2))) float v2f;
typedef __attribute__((ext_vector_type(8))) float v8f;

#define N_NODES 100000
#define E_EDGES 1600000
#define DIM     128
#define DIM4    (DIM / 4)
#define APAD    4              // LDS row pad: stride 132 -> conflict-free b64 reads
#define ASTRIDE (DIM + APAD)

// ---------------------------------------------------------------------------
// Degree / normalization helpers
// ---------------------------------------------------------------------------
__global__ void deg_init_kernel(float* __restrict__ deg, int n) {
    int i = blockIdx.x * blockDim.x + threadIdx.x;
    if (i < n) deg[i] = 1.0f;  // self-loop contributes 1 to every node
}

__global__ void deg_count_kernel(const int* __restrict__ dst,
                                 float* __restrict__ deg, int e) {
    int i = blockIdx.x * blockDim.x + threadIdx.x;
    if (i < e) atomicAdd(&deg[dst[i]], 1.0f);
}

__global__ void deg_rsqrt_kernel(float* __restrict__ deg, int n) {
    int i = blockIdx.x * blockDim.x + threadIdx.x;
    if (i < n) deg[i] = rsqrtf(deg[i]);  // deg >= 1 always (self-loop)
}

// ---------------------------------------------------------------------------
// GEMM: out[n][d] = sum_k in[n][k] * W[d][k]   (i.e. in @ W^T), fp32 WMMA.
// Block = 256 threads (8 waves) -> 16 rows x 128 cols of out.
// The 16x128 A tile is staged in LDS once (coalesced), then each wave pulls
// its A fragments via ds_load_b64 (padded stride 132 -> 64 distinct banks).
// B fragments come straight from W in global (64 KB, L2/WGP$-resident).
//   A(16x4 f32): lane L<16 holds {K0,K1} of row L; lane L>=16 holds {K2,K3}.
//   B(4x16 f32): same pattern over cols; B[k][n] = W[n][k] -> read W rows.
//   C/D(16x16 f32): col = lane%16, vgpr i -> row i + 8*(lane>=16).
// ---------------------------------------------------------------------------
__global__ __launch_bounds__(256) void gemm_xWT_wmma_kernel(
    const float* __restrict__ in, const float* __restrict__ W,
    float* __restrict__ out, int nrows)
{
    __shared__ float As[16 * ASTRIDE];  // 8448 B

    const int tid     = threadIdx.x;
    const int lane    = tid & 31;
    const int wave    = tid >> 5;
    const int rowBase = blockIdx.x * 16;
    const int colBase = wave * 16;
    if (rowBase >= nrows) return;  // uniform per block; N is a multiple of 16

    // Cooperative, fully coalesced copy of the 16x128 A tile into LDS.
    const float* ablk = in + (size_t)rowBase * DIM;
#pragma unroll
    for (int i = 0; i < (16 * DIM) / 256; ++i) {
        int idx = tid + i * 256;
        int r = idx >> 7;        // idx / DIM
        int c = idx & (DIM - 1); // idx % DIM
        As[r * ASTRIDE + c] = ablk[idx];
    }
    __syncthreads();

    const int half = lane >> 4;   // 0 for lanes 0-15, 1 for lanes 16-31
    const int l16  = lane & 15;
    const int klo  = half * 2;    // K offset within a K4 step: 0 or 2

    const float* aldsrow = &As[l16 * ASTRIDE + klo];
    const float* brow    = W + (size_t)(colBase + l16) * DIM + klo;

    v8f acc = {};
#pragma unroll
    for (int k0 = 0; k0 < DIM; k0 += 4) {
        float2 av = *(const float2*)(aldsrow + k0);  // ds_load_b64, bank-clean
        float2 bv = *(const float2*)(brow + k0);
        v2f a; a.x = av.x; a.y = av.y;
        v2f b; b.x = bv.x; b.y = bv.y;
        acc = __builtin_amdgcn_wmma_f32_16x16x4_f32(
            /*neg_a=*/false, a, /*neg_b=*/false, b,
            /*c_mod=*/(short)0, acc, /*reuse_a=*/false, /*reuse_b=*/false);
    }

    float* orow = out + (size_t)(rowBase + half * 8) * DIM + colBase + l16;
#pragma unroll
    for (int i = 0; i < 8; ++i)
        orow[(size_t)i * DIM] = acc[i];
}

// ---------------------------------------------------------------------------
// out[i][:] = bias + perturb[i][:] + h[i][:] * dinv[i]^2   (self-loop term)
// ---------------------------------------------------------------------------
__global__ void fuse_self_kernel(const float* __restrict__ h,
                                 const float* __restrict__ bias,
                                 const float* __restrict__ pert,
                                 const float* __restrict__ dinv,
                                 float* __restrict__ out, int n)
{
    int t = blockIdx.x * blockDim.x + threadIdx.x;
    if (t >= n * DIM4) return;
    int node = t / DIM4;
    int c    = t % DIM4;
    float s = dinv[node];
    s = s * s;
    float4 hv = ((const float4*)h)[t];
    float4 pv = ((const float4*)pert)[t];
    float4 bv = ((const float4*)bias)[c];
    float4 o;
    o.x = bv.x + pv.x + hv.x * s;
    o.y = bv.y + pv.y + hv.y * s;
    o.z = bv.z + pv.z + hv.z * s;
    o.w = bv.w + pv.w + hv.w * s;
    ((float4*)out)[t] = o;
}

// ---------------------------------------------------------------------------
// Edge scatter: out[dst] += h[src] * dinv[src]*dinv[dst]
// One wave per edge: edge id derived from (blockIdx, wave) so src/dst/dinv
// lookups are wave-uniform (scalarizable); 32 lanes x float4 = 128 features,
// coalesced 512B row read, coalesced atomics on the destination row.
// ---------------------------------------------------------------------------
__global__ __launch_bounds__(256) void scatter_edges_kernel(
    const float* __restrict__ h,
    const int* __restrict__ src,
    const int* __restrict__ dst,
    const float* __restrict__ dinv,
    float* __restrict__ out, int e)
{
    const int wave = threadIdx.x >> 5;   // wave-uniform on wave32
    const int lane = threadIdx.x & 31;
    const long long edge = (long long)blockIdx.x * 8 + wave;
    if (edge >= e) return;               // uniform within the wave
    int s = src[edge];
    int d = dst[edge];
    float w = dinv[s] * dinv[d];
    float4 hv = ((const float4*)h)[(size_t)s * DIM4 + lane];
    float* o = out + (size_t)d * DIM + lane * 4;
    atomicAdd(o + 0, hv.x * w);
    atomicAdd(o + 1, hv.y * w);
    atomicAdd(o + 2, hv.z * w);
    atomicAdd(o + 3, hv.w * w);
}

// ---------------------------------------------------------------------------
// Launcher
// inputs: x, edge_index(2,E), perturb_first, perturb_last, W1, b1, W2, b2
// ---------------------------------------------------------------------------
extern "C" void kernel_launch(void* const* d_in, const int* in_sizes, int n_in,
                              void* d_out, int out_size, void* d_ws, size_t ws_size,
                              hipStream_t stream) {
    const float* x     = (const float*)d_in[0];
    const int*   eidx  = (const int*)d_in[1];
    const float* pertA = (const float*)d_in[2];
    const float* pertB = (const float*)d_in[3];
    const float* W1    = (const float*)d_in[4];
    const float* b1    = (const float*)d_in[5];
    const float* W2    = (const float*)d_in[6];
    const float* b2    = (const float*)d_in[7];
    float* out = (float*)d_out;

    const int N = N_NODES;
    const int E = in_sizes[1] / 2;  // edge_index is (2,E) row-major
    const int* srcIdx = eidx;       // edge_index[0]
    const int* dstIdx = eidx + E;   // edge_index[1]

    // Workspace carve-up: dinv[N] | h[N*DIM] | aggr[N*DIM]
    float* dinv = (float*)d_ws;
    float* h    = dinv + N;
    float* aggr = h + (size_t)N * DIM;

    const int T = 256;
    dim3 blk(T);

    // --- normalization ---
    deg_init_kernel<<<dim3((N + T - 1) / T), blk, 0, stream>>>(dinv, N);
    deg_count_kernel<<<dim3((E + T - 1) / T), blk, 0, stream>>>(dstIdx, dinv, E);
    deg_rsqrt_kernel<<<dim3((N + T - 1) / T), blk, 0, stream>>>(dinv, N);

    const int gemmBlocks = N / 16;                 // 6250, N % 16 == 0
    const int edgeBlocks = (E + 7) / 8;            // one wave (of 8/block) per edge
    const int nodeBlocks = (int)(((long long)N * DIM4 + T - 1) / T);

    // --- layer 1 ---
    gemm_xWT_wmma_kernel<<<dim3(gemmBlocks), blk, 0, stream>>>(x, W1, h, N);
    fuse_self_kernel<<<dim3(nodeBlocks), blk, 0, stream>>>(h, b1, pertA, dinv, aggr, N);
    scatter_edges_kernel<<<dim3(edgeBlocks), blk, 0, stream>>>(h, srcIdx, dstIdx, dinv, aggr, E);

    // --- layer 2 ---
    gemm_xWT_wmma_kernel<<<dim3(gemmBlocks), blk, 0, stream>>>(aggr, W2, h, N);
    fuse_self_kernel<<<dim3(nodeBlocks), blk, 0, stream>>>(h, b2, pertB, dinv, out, N);
    scatter_edges_kernel<<<dim3(edgeBlocks), blk, 0, stream>>>(h, srcIdx, dstIdx, dinv, out, E);
}